// DAGDecoder_50268297232754
// MI455X (gfx1250) — compile-verified
//
#include <hip/hip_runtime.h>
#include <hip/hip_bf16.h>

#define B_LATENT 256
#define HIDDEN   512
#define NTYPES   16
#define NNODES   128

typedef float v2f __attribute__((ext_vector_type(2)));
typedef float v8f __attribute__((ext_vector_type(8)));

// C[M,N] = act(A[M,K] @ B[K,N] (+ B[K+K,N] second half if DUALB) + bias[N])
// One wave32 computes one 16x16 tile with V_WMMA_F32_16X16X4_F32 (exact f32).
template <bool RELU, bool DUALB>
__global__ __launch_bounds__(128) void gemm_wmma_f32(
    const float* __restrict__ A, const float* __restrict__ Bm,
    const float* __restrict__ bias, float* __restrict__ C,
    int M, int N, int K)
{
    const int wave    = blockIdx.x * (blockDim.x >> 5) + (threadIdx.x >> 5);
    const int tiles_n = N >> 4;
    const int tm   = wave / tiles_n;
    const int tn   = wave % tiles_n;
    const int lane = threadIdx.x & 31;
    const int half = lane >> 4;   // 0 or 1
    const int lid  = lane & 15;
    const int row  = (tm << 4) + lid;   // A-matrix row this lane feeds
    const int col  = (tn << 4) + lid;   // B-matrix column this lane feeds
    const long dualOff = (long)K * N;   // offset to W_e1's second 512 rows

    v8f c = {0.f, 0.f, 0.f, 0.f, 0.f, 0.f, 0.f, 0.f};
    const float* __restrict__ arow = A + (long)row * K;

    for (int k0 = 0; k0 < K; k0 += 4) {
        const int kb = k0 + (half << 1);                 // per-lane K base
        v2f a = *(const v2f*)(arow + kb);                // A[row, kb..kb+1]
        float b0 = Bm[(long)kb * N + col];
        float b1 = Bm[(long)(kb + 1) * N + col];
        if (DUALB) {
            b0 += Bm[(long)kb * N + col + dualOff];
            b1 += Bm[(long)(kb + 1) * N + col + dualOff];
        }
        v2f b = { b0, b1 };
        // (neg_a, A, neg_b, B, c_mod, C, reuse_a, reuse_b)
        c = __builtin_amdgcn_wmma_f32_16x16x4_f32(
                false, a, false, b, (short)0, c, false, false);
    }

    const float bc = bias[col];
#pragma unroll
    for (int v = 0; v < 8; ++v) {
        const int m = (tm << 4) + v + (half << 3);       // D layout: M = v + 8*half
        float val = c[v] + bc;
        if (RELU) val = fmaxf(val, 0.f);
        C[(long)m * N + col] = val;
    }
}

// p[row] = sigmoid(dot(T[row,:], w) + b2)  — one wave32 per row
__global__ __launch_bounds__(256) void gemv_sigmoid(
    const float* __restrict__ T, const float* __restrict__ w,
    const float* __restrict__ b2, float* __restrict__ p, int K)
{
    const int row  = blockIdx.x * (blockDim.x >> 5) + (threadIdx.x >> 5);
    const int lane = threadIdx.x & 31;
    float s = 0.f;
    for (int k = lane; k < K; k += 32)
        s += T[(long)row * K + k] * w[k];
#pragma unroll
    for (int off = 16; off > 0; off >>= 1)
        s += __shfl_down(s, off, 32);
    if (lane == 0)
        p[row] = 1.f / (1.f + expf(-(s + b2[0])));
}

// out[b,n,0:16] = nl[b,0:16] for all n; float4-vectorized store stream
__global__ __launch_bounds__(256) void bcast_node_logits(
    const float* __restrict__ nl, float* __restrict__ out, int total4)
{
    const int idx = blockIdx.x * blockDim.x + threadIdx.x;  // over B*N*4
    if (idx >= total4) return;
    const int t4 = idx & 3;
    const int b  = (idx >> 2) / NNODES;
    const float4 v = *(const float4*)(nl + b * NTYPES + (t4 << 2));
    *(float4*)(out + (long)idx * 4) = v;
}

// out[b,i,j] = (j < i) ? p[b] : 0 ; float4-vectorized store stream
__global__ __launch_bounds__(256) void fill_edge_probs(
    const float* __restrict__ p, float* __restrict__ out, int total4)
{
    const int idx = blockIdx.x * blockDim.x + threadIdx.x;  // over B*128*32
    if (idx >= total4) return;
    const int j  = (idx & (NNODES / 4 - 1)) << 2;   // 0..124 step 4
    const int i  = (idx >> 5) & (NNODES - 1);
    const int b  = idx >> 12;
    const float pb = p[b];
    float4 v;
    v.x = (j + 0 < i) ? pb : 0.f;
    v.y = (j + 1 < i) ? pb : 0.f;
    v.z = (j + 2 < i) ? pb : 0.f;
    v.w = (j + 3 < i) ? pb : 0.f;
    *(float4*)(out + (long)idx * 4) = v;
}

extern "C" void kernel_launch(void* const* d_in, const int* in_sizes, int n_in,
                              void* d_out, int out_size, void* d_ws, size_t ws_size,
                              hipStream_t stream)
{
    const float* z      = (const float*)d_in[0];
    // d_in[1] = num_nodes (int scalar); fixed at 128 by out_size, baked in.
    const float* W_z    = (const float*)d_in[2];
    const float* b_z    = (const float*)d_in[3];
    const float* W_node = (const float*)d_in[4];
    const float* b_node = (const float*)d_in[5];
    const float* W_e1   = (const float*)d_in[6];
    const float* b_e1   = (const float*)d_in[7];
    const float* W_e2   = (const float*)d_in[8];
    const float* b_e2   = (const float*)d_in[9];

    const int B = in_sizes[0] / B_LATENT;   // 1024

    float* ws  = (float*)d_ws;
    float* h   = ws;                          // [B, 512]
    float* t   = h  + (size_t)B * HIDDEN;     // [B, 512]
    float* pv  = t  + (size_t)B * HIDDEN;     // [B]
    float* nl  = pv + (size_t)B;              // [B, 16]

    float* out_nl = (float*)d_out;                              // [B, 128, 16]
    float* out_ep = out_nl + (size_t)B * NNODES * NTYPES;       // [B, 128, 128]

    // 1) h = relu(z @ W_z + b_z)                [B,512], K=256
    {
        const int waves = (B / 16) * (HIDDEN / 16);
        gemm_wmma_f32<true, false><<<waves / 4, 128, 0, stream>>>(
            z, W_z, b_z, h, B, HIDDEN, B_LATENT);
    }
    // 2) t = relu(h @ (W_e1_lo + W_e1_hi) + b_e1)   [B,512], K=512
    {
        const int waves = (B / 16) * (HIDDEN / 16);
        gemm_wmma_f32<true, true><<<waves / 4, 128, 0, stream>>>(
            h, W_e1, b_e1, t, B, HIDDEN, HIDDEN);
    }
    // 3) nl = h @ W_node + b_node               [B,16], K=512
    {
        const int waves = (B / 16) * (NTYPES / 16);
        gemm_wmma_f32<false, false><<<waves / 4, 128, 0, stream>>>(
            h, W_node, b_node, nl, B, NTYPES, HIDDEN);
    }
    // 4) p = sigmoid(t @ W_e2 + b_e2)           [B]
    gemv_sigmoid<<<B / 8, 256, 0, stream>>>(t, W_e2, b_e2, pv, HIDDEN);

    // 5) broadcast node logits across node dim  (8 MB stores)
    {
        const int total4 = B * NNODES * (NTYPES / 4);
        bcast_node_logits<<<(total4 + 255) / 256, 256, 0, stream>>>(nl, out_nl, total4);
    }
    // 6) p scattered on strict lower triangle   (67 MB stores)
    {
        const int total4 = B * NNODES * (NNODES / 4);
        fill_edge_probs<<<(total4 + 255) / 256, 256, 0, stream>>>(pv, out_ep, total4);
    }
}